// GCNEncoder_27642409517639
// MI455X (gfx1250) — compile-verified
//
#include <hip/hip_runtime.h>

#define NNODES 51200
#define NEDGES 819200
#define POOLH  32
#define NODES_PER_POOL (NNODES / POOLH)   // 1600

typedef float v2f __attribute__((ext_vector_type(2)));
typedef float v8f __attribute__((ext_vector_type(8)));

// ---------------- degree / norm ----------------
__global__ void k_init_deg(float* deg, int n) {
    int i = blockIdx.x * blockDim.x + threadIdx.x;
    if (i < n) deg[i] = 1.0f;                      // self-loop contribution
}

__global__ void k_count_deg(const int* __restrict__ dst, float* deg, int e) {
    int i = blockIdx.x * blockDim.x + threadIdx.x;
    if (i < e) atomicAdd(&deg[dst[i]], 1.0f);
}

__global__ void k_rsqrt_inplace(float* d, int n) {
    int i = blockIdx.x * blockDim.x + threadIdx.x;
    if (i < n) d[i] = rsqrtf(d[i]);                // deg >= 1 always
}

// ---------------- f32 WMMA GEMM: C(MxN) = A(MxK) @ B(KxN), one 16x16 tile per wave ----------------
template<int KDIM>
__global__ void k_gemm_wmma(const float* __restrict__ A, const float* __restrict__ B,
                            float* __restrict__ C, int M, int Ncols) {
    int wave  = (int)((blockIdx.x * blockDim.x + threadIdx.x) >> 5);   // wave-uniform
    int lane  = threadIdx.x & 31;
    int tcols = Ncols >> 4;
    int trow  = wave / tcols;
    int tcol  = wave % tcols;
    if ((trow << 4) >= M) return;                  // wave-uniform exit: EXEC all-ones for WMMA
    int half = lane >> 4;                          // 0: K pair {0,1}, 1: K pair {2,3}
    int m    = lane & 15;

    const float* Arow = A + (size_t)(trow * 16 + m) * KDIM;
    const float* Bcol = B + (tcol * 16 + m);

    v8f acc = {0.f, 0.f, 0.f, 0.f, 0.f, 0.f, 0.f, 0.f};
#pragma unroll
    for (int k0 = 0; k0 < KDIM; k0 += 4) {
        int ka = k0 + half * 2;
        v2f a, b;
        a.x = Arow[ka];
        a.y = Arow[ka + 1];
        b.x = Bcol[(size_t)ka * Ncols];
        b.y = Bcol[(size_t)(ka + 1) * Ncols];
        // 8 args: (neg_a, A, neg_b, B, c_mod, C, reuse_a, reuse_b)
        acc = __builtin_amdgcn_wmma_f32_16x16x4_f32(false, a, false, b,
                                                    (short)0, acc, false, false);
    }
    // C/D layout: VGPR r -> rows {r, r+8}; lanes 0-15 cols, 16-31 cols (M+8)
    float* Crow = C + (size_t)(trow * 16 + half * 8) * Ncols + tcol * 16 + m;
#pragma unroll
    for (int r = 0; r < 8; ++r)
        Crow[(size_t)r * Ncols] = acc[r];
}

// ---------------- normalized edge scatter-add ----------------
template<int F>
__global__ void k_scatter(const int* __restrict__ src, const int* __restrict__ dst,
                          const float* __restrict__ dinv, const float* __restrict__ xw,
                          float* out, int e) {
    long long tid = (long long)blockIdx.x * blockDim.x + threadIdx.x;
    int edge = (int)(tid / F);
    int f    = (int)(tid % F);
    if (edge >= e) return;
    int s = src[edge], d = dst[edge];
    float norm = dinv[s] * dinv[d];
    atomicAdd(&out[(size_t)d * F + f], norm * xw[(size_t)s * F + f]);
}

// ---------------- fused self-loop + bias + ReLU (layer 1), in place on h ----------------
__global__ void k_bias_relu_self(float* __restrict__ h, const float* __restrict__ xw,
                                 const float* __restrict__ dinv, const float* __restrict__ b,
                                 int total) {                       // total = N*64
    int i = blockIdx.x * blockDim.x + threadIdx.x;
    if (i >= total) return;
    int node = i >> 6;
    int f    = i & 63;
    float di = dinv[node];
    float v  = h[i] + di * di * xw[i] + b[f];
    h[i] = v > 0.f ? v : 0.f;
}

// ---------------- pooled output: init with b2, then accumulate block means ----------------
__global__ void k_init_out(float* out, const float* __restrict__ b2) {
    int i = blockIdx.x * blockDim.x + threadIdx.x;  // 32*128 = 4096 threads
    out[i] = b2[i & 127];
}

__global__ void k_pool(const float* __restrict__ hs, const float* __restrict__ xw,
                       const float* __restrict__ dinv, float* out) {
    int p  = blockIdx.x;              // pool row 0..31
    int s  = blockIdx.y;              // sub-chunk 0..15
    int f  = threadIdx.x;             // feature 0..127
    int i0 = p * NODES_PER_POOL + s * (NODES_PER_POOL / 16);
    float sum = 0.f;
    for (int i = i0; i < i0 + NODES_PER_POOL / 16; ++i) {
        float di = dinv[i];
        sum += hs[(size_t)i * 128 + f] + di * di * xw[(size_t)i * 128 + f];
    }
    atomicAdd(&out[p * 128 + f], sum * (1.0f / (float)NODES_PER_POOL));
}

extern "C" void kernel_launch(void* const* d_in, const int* in_sizes, int n_in,
                              void* d_out, int out_size, void* d_ws, size_t ws_size,
                              hipStream_t stream) {
    (void)in_sizes; (void)n_in; (void)out_size; (void)ws_size;

    const float* x  = (const float*)d_in[0];   // (N,16)
    const int*   ei = (const int*)d_in[1];     // (2,E)
    const float* W1 = (const float*)d_in[2];   // (16,64)
    const float* b1 = (const float*)d_in[3];   // (64,)
    const float* W2 = (const float*)d_in[4];   // (64,128)
    const float* b2 = (const float*)d_in[5];   // (128,)
    float* out = (float*)d_out;                // (1,32,128)

    const int N = NNODES, E = NEDGES;
    const int* srcp = ei;
    const int* dstp = ei + E;

    // workspace layout: dinv (N) | R1 (N*128) | R2 (N*128)  ~= 52.6 MB
    float* dinv = (float*)d_ws;
    float* R1   = dinv + N;
    float* R2   = R1 + (size_t)N * 128;

    // degrees -> D^{-1/2}
    k_init_deg<<<(N + 255) / 256, 256, 0, stream>>>(dinv, N);
    k_count_deg<<<(E + 255) / 256, 256, 0, stream>>>(dstp, dinv, E);
    k_rsqrt_inplace<<<(N + 255) / 256, 256, 0, stream>>>(dinv, N);

    // GEMM1: R1[:, :64] = x @ W1  (f32 WMMA, 3200x4 tiles, 4 waves/block)
    {
        int waves = (N / 16) * (64 / 16);
        k_gemm_wmma<16><<<waves / 4, 128, 0, stream>>>(x, W1, R1, N, 64);
    }

    // layer-1 scatter into R2[:, :64]
    hipMemsetAsync(R2, 0, (size_t)N * 64 * sizeof(float), stream);
    {
        long long threads = (long long)E * 64;
        k_scatter<64><<<(int)((threads + 255) / 256), 256, 0, stream>>>(srcp, dstp, dinv, R1, R2, E);
    }
    // h1 = relu(scatter + dinv^2 * xw1 + b1), in place on R2
    k_bias_relu_self<<<(N * 64 + 255) / 256, 256, 0, stream>>>(R2, R1, dinv, b1, N * 64);

    // GEMM2: R1 = h1 @ W2  (3200x8 tiles)
    {
        int waves = (N / 16) * (128 / 16);
        k_gemm_wmma<64><<<waves / 4, 128, 0, stream>>>(R2, W2, R1, N, 128);
    }

    // layer-2 scatter into R2 (full N*128)
    hipMemsetAsync(R2, 0, (size_t)N * 128 * sizeof(float), stream);
    {
        long long threads = (long long)E * 128;
        k_scatter<128><<<(int)((threads + 255) / 256), 256, 0, stream>>>(srcp, dstp, dinv, R1, R2, E);
    }

    // pooled output = b2 + mean over node blocks of (scatter + dinv^2 * xw2)
    k_init_out<<<(POOLH * 128) / 256, 256, 0, stream>>>(out, b2);
    k_pool<<<dim3(POOLH, 16, 1), 128, 0, stream>>>(R2, R1, dinv, out);
}